// Head_82592221102487
// MI455X (gfx1250) — compile-verified
//
#include <hip/hip_runtime.h>
#include <hip/hip_bf16.h>

#define B_DIM 8
#define T_DIM 2048
#define E_DIM 1024
#define D_DIM 64          // head size
#define NEG_BIG (-1.0e30f)

typedef __attribute__((ext_vector_type(2))) float v2f;
typedef __attribute__((ext_vector_type(8))) float v8f;

// ---------------------------------------------------------------------------
// Stage 1: q/k/v projections.  One wave computes one 16x16 tile of one of the
// three [B*T, 64] outputs using V_WMMA_F32_16X16X4_F32 (exact f32 math).
// Tiles: mat(3) x mtile(B*T/16 = 1024) x ntile(D/16 = 4) = 12288 tiles.
// ---------------------------------------------------------------------------
__global__ void __launch_bounds__(256)
proj_kernel(const float* __restrict__ x,
            const float* __restrict__ Wk,
            const float* __restrict__ Wq,
            const float* __restrict__ Wv,
            float* __restrict__ kq,
            float* __restrict__ qq,
            float* __restrict__ vq)
{
    const int wave = threadIdx.x >> 5;
    const int lane = threadIdx.x & 31;
    const int tile = blockIdx.x * 8 + wave;      // 0..12287

    const int ntile = tile & 3;                  // D tile
    const int mtile = (tile >> 2) & 1023;        // B*T tile
    const int mat   = tile >> 12;                // 0=k, 1=q, 2=v

    const float* W   = (mat == 0) ? Wk : (mat == 1) ? Wq : Wv;
    float*       out = (mat == 0) ? kq : (mat == 1) ? qq : vq;

    const int m16 = lane & 15;                   // A: M index / B: N index
    const int hi  = lane >> 4;                   // K-half selector

    const float* xrow = x + (size_t)(mtile * 16 + m16) * E_DIM + 2 * hi;
    const float* wrow = W + (size_t)(ntile * 16 + m16) * E_DIM + 2 * hi;

    v8f acc = {};
#pragma unroll 4
    for (int e0 = 0; e0 < E_DIM; e0 += 4) {
        // A[m][k] = x[mtile*16+m][e0+k], lane holds k = 2*hi + {0,1}
        v2f a = *(const v2f*)(xrow + e0);
        // B[k][n] = W[ntile*16+n][e0+k]
        v2f b = *(const v2f*)(wrow + e0);
        acc = __builtin_amdgcn_wmma_f32_16x16x4_f32(
            false, a, false, b, (short)0, acc, false, false);
    }

    // C/D layout: VGPR r -> M = r + 8*hi, N = lane&15
    const int Mbase = mtile * 16 + 8 * hi;
    const int N     = ntile * 16 + m16;
#pragma unroll
    for (int r = 0; r < 8; ++r)
        out[(size_t)(Mbase + r) * D_DIM + N] = acc[r];
}

// ---------------------------------------------------------------------------
// Stage 2: causal flash attention with online softmax.
// One wave owns one 16-row query tile (B * T/16 = 1024 tiles).
// ---------------------------------------------------------------------------
__global__ void __launch_bounds__(256)
attn_kernel(const float* __restrict__ q,
            const float* __restrict__ k,
            const float* __restrict__ v,
            float* __restrict__ out)
{
    __shared__ float lds[8 * 256];               // 1 KB P-staging per wave

    const int wave = threadIdx.x >> 5;
    const int lane = threadIdx.x & 31;
    float* pbuf = lds + wave * 256;

    const int tile = blockIdx.x * 8 + wave;      // 0..1023
    const int b    = tile >> 7;                  // T/16 = 128 tiles per batch
    const int qt   = tile & 127;

    const int m16 = lane & 15;
    const int hi  = lane >> 4;
    const float scale = 0.125f;                  // 1/sqrt(64)

    const float* qb = q + (size_t)b * T_DIM * D_DIM;
    const float* kb = k + (size_t)b * T_DIM * D_DIM;
    const float* vb = v + (size_t)b * T_DIM * D_DIM;

    // Preload Q A-fragments: a[kk] holds Q[qt*16+m16][kk*4 + 2*hi + {0,1}]
    v2f qa[16];
    {
        const float* qrow = qb + (size_t)(qt * 16 + m16) * D_DIM + 2 * hi;
#pragma unroll
        for (int kk = 0; kk < 16; ++kk)
            qa[kk] = *(const v2f*)(qrow + kk * 4);
    }

    v8f o0 = {}, o1 = {}, o2 = {}, o3 = {};      // O tile: 16 x 64 (4 N-tiles)
    float mrow[8], lrow[8];
#pragma unroll
    for (int r = 0; r < 8; ++r) { mrow[r] = NEG_BIG; lrow[r] = 0.0f; }

    for (int j = 0; j <= qt; ++j) {
        // ---- S = Q * K^T : 16 WMMAs over D=64 ----
        v8f s = {};
        const float* krow = kb + (size_t)(j * 16 + m16) * D_DIM + 2 * hi;
#pragma unroll
        for (int kk = 0; kk < 16; ++kk) {
            v2f bf = *(const v2f*)(krow + kk * 4);   // B[kd][n] = K[j*16+n][kd]
            s = __builtin_amdgcn_wmma_f32_16x16x4_f32(
                false, qa[kk], false, bf, (short)0, s, false, false);
        }

        // ---- scale, causal mask, online softmax (rows span 16-lane halves) --
        const int nglob = j * 16 + m16;
        float alpha[8];
#pragma unroll
        for (int r = 0; r < 8; ++r) {
            const int mglob = qt * 16 + r + 8 * hi;
            float sv = s[r] * scale;
            if (nglob > mglob) sv = NEG_BIG;

            float mx = sv;
            mx = fmaxf(mx, __shfl_xor(mx, 1, 32));
            mx = fmaxf(mx, __shfl_xor(mx, 2, 32));
            mx = fmaxf(mx, __shfl_xor(mx, 4, 32));
            mx = fmaxf(mx, __shfl_xor(mx, 8, 32));
            const float mnew = fmaxf(mrow[r], mx);

            const float p = __expf(sv - mnew);
            float sum = p;
            sum += __shfl_xor(sum, 1, 32);
            sum += __shfl_xor(sum, 2, 32);
            sum += __shfl_xor(sum, 4, 32);
            sum += __shfl_xor(sum, 8, 32);

            const float a = __expf(mrow[r] - mnew);
            lrow[r] = lrow[r] * a + sum;
            mrow[r] = mnew;
            alpha[r] = a;
            s[r] = p;                             // s now holds P (C layout)
        }
#pragma unroll
        for (int r = 0; r < 8; ++r) {
            o0[r] *= alpha[r]; o1[r] *= alpha[r];
            o2[r] *= alpha[r]; o3[r] *= alpha[r];
        }

        // ---- P: C-layout -> LDS -> A-layout ----
#pragma unroll
        for (int r = 0; r < 8; ++r)
            pbuf[(r + 8 * hi) * 16 + m16] = s[r];
        __asm__ volatile("s_wait_dscnt 0" ::: "memory");

        // ---- O += P * V : 4 K-chunks x 4 D-tiles = 16 WMMAs ----
#pragma unroll
        for (int kk = 0; kk < 4; ++kk) {
            v2f pa = *(const v2f*)(pbuf + m16 * 16 + kk * 4 + 2 * hi);
            // B[kd][n] = V[j*16 + kd][dt*16 + n], kd = kk*4 + 2*hi + {0,1}
            const float* vr = vb + (size_t)(j * 16 + kk * 4 + 2 * hi) * D_DIM + m16;
            v2f b0; b0.x = vr[0];        b0.y = vr[0 + D_DIM];
            v2f b1; b1.x = vr[16];       b1.y = vr[16 + D_DIM];
            v2f b2; b2.x = vr[32];       b2.y = vr[32 + D_DIM];
            v2f b3; b3.x = vr[48];       b3.y = vr[48 + D_DIM];
            o0 = __builtin_amdgcn_wmma_f32_16x16x4_f32(false, pa, false, b0, (short)0, o0, false, false);
            o1 = __builtin_amdgcn_wmma_f32_16x16x4_f32(false, pa, false, b1, (short)0, o1, false, false);
            o2 = __builtin_amdgcn_wmma_f32_16x16x4_f32(false, pa, false, b2, (short)0, o2, false, false);
            o3 = __builtin_amdgcn_wmma_f32_16x16x4_f32(false, pa, false, b3, (short)0, o3, false, false);
        }
    }

    // ---- normalize and store:  out[b, qt*16 + M, :] ----
#pragma unroll
    for (int r = 0; r < 8; ++r) {
        const float inv = 1.0f / lrow[r];
        float* orow = out + (size_t)(((size_t)b * T_DIM) + qt * 16 + r + 8 * hi) * D_DIM + m16;
        orow[0]  = o0[r] * inv;
        orow[16] = o1[r] * inv;
        orow[32] = o2[r] * inv;
        orow[48] = o3[r] * inv;
    }
}

// ---------------------------------------------------------------------------
extern "C" void kernel_launch(void* const* d_in, const int* in_sizes, int n_in,
                              void* d_out, int out_size, void* d_ws, size_t ws_size,
                              hipStream_t stream)
{
    const float* x  = (const float*)d_in[0];
    const float* Wk = (const float*)d_in[1];
    const float* Wq = (const float*)d_in[2];
    const float* Wv = (const float*)d_in[3];
    float* out = (float*)d_out;

    const size_t btd = (size_t)B_DIM * T_DIM * D_DIM;   // 1,048,576 floats
    float* kq = (float*)d_ws;
    float* qq = kq + btd;
    float* vq = qq + btd;

    // 3 * 1024 * 4 tiles, 8 waves (tiles) per 256-thread block
    proj_kernel<<<1536, 256, 0, stream>>>(x, Wk, Wq, Wv, kq, qq, vq);
    // 1024 query tiles, 8 per block
    attn_kernel<<<128, 256, 0, stream>>>(qq, kq, vq, out);
}